// GCNEncoder_31370441130066
// MI455X (gfx1250) — compile-verified
//
#include <hip/hip_runtime.h>
#include <hip/hip_bf16.h>
#include <stdint.h>

typedef __attribute__((ext_vector_type(2))) float v2f;
typedef __attribute__((ext_vector_type(8))) float v8f;

// ---------------------------------------------------------------------------
// Utility: zero a float buffer (grid-stride)
// ---------------------------------------------------------------------------
__global__ void zero_kernel(float* __restrict__ p, int n) {
  int i = blockIdx.x * blockDim.x + threadIdx.x;
  int stride = gridDim.x * blockDim.x;
  for (; i < n; i += stride) p[i] = 0.0f;
}

// ---------------------------------------------------------------------------
// Degree: deg[dst] += ew  over E edges, plus +1.0 self-loop per node
// ---------------------------------------------------------------------------
__global__ void degree_kernel(const long long* __restrict__ dstI,
                              const float* __restrict__ ew,
                              float* __restrict__ deg, int E, int N) {
  int i = blockIdx.x * blockDim.x + threadIdx.x;
  int total = E + N;
  if (i >= total) return;
  if (i < E) {
    atomicAdd(&deg[(int)dstI[i]], ew[i]);
  } else {
    atomicAdd(&deg[i - E], 1.0f);
  }
}

// dinv = deg > 0 ? 1/sqrt(deg) : 0   (full precision to match jax.lax.rsqrt)
__global__ void dinv_kernel(const float* __restrict__ deg,
                            float* __restrict__ dinv, int N) {
  int i = blockIdx.x * blockDim.x + threadIdx.x;
  if (i >= N) return;
  float d = deg[i];
  dinv[i] = (d > 0.0f) ? (1.0f / sqrtf(d)) : 0.0f;
}

// ---------------------------------------------------------------------------
// Dense transform C[M x NCOLS] = A[M x 64] * B[64 x NCOLS] via
// V_WMMA_F32_16X16X4_F32. One wave per 16x16 output tile; K=64 as 16 steps.
//
// f32 WMMA fragment layout (ISA 7.12.2):
//   A 16x4 : lanes 0-15 hold M=lane, K = {0,1} in v0,v1 ; lanes 16-31 K={2,3}
//   B 4x16 : lanes 0-15 hold N=lane, K = {0,1} in v0,v1 ; lanes 16-31 K={2,3}
//   C 16x16: VGPR r -> row (r + 8*(lane>>4)), col (lane&15)
// ---------------------------------------------------------------------------
template <int NCOLS>
__global__ void gemm_wmma_kernel(const float* __restrict__ A,
                                 const float* __restrict__ B,
                                 float* __restrict__ C, int M) {
  const int lane = threadIdx.x & 31;
  const int wave = threadIdx.x >> 5;
  const int tile = blockIdx.x * (blockDim.x >> 5) + wave;
  const int colTiles = NCOLS / 16;
  const int rowTile = tile / colTiles;
  const int colTile = tile - rowTile * colTiles;
  const int row0 = rowTile * 16;
  if (row0 >= M) return;  // M % 16 == 0, so a live tile is always full
  const int col0 = colTile * 16;

  const int m = row0 + (lane & 15);         // A row for this lane
  const int n = col0 + (lane & 15);         // B/C column for this lane
  const int khalf = (lane >> 4) << 1;       // 0 for lanes 0-15, 2 for 16-31

  v8f acc = {};
#pragma unroll
  for (int k = 0; k < 64; k += 4) {
    // A fragment: two consecutive K values for this lane's row
    v2f a = *(const v2f*)(&A[(size_t)m * 64 + k + khalf]);
    // B fragment: two K-rows of this lane's column (stride NCOLS)
    v2f b;
    b.x = B[(size_t)(k + khalf + 0) * NCOLS + n];
    b.y = B[(size_t)(k + khalf + 1) * NCOLS + n];
    acc = __builtin_amdgcn_wmma_f32_16x16x4_f32(
        /*neg_a=*/false, a, /*neg_b=*/false, b,
        /*c_mod=*/(short)0, acc, /*reuse_a=*/false, /*reuse_b=*/false);
  }

  const int rowBase = row0 + ((lane >> 4) << 3);
#pragma unroll
  for (int r = 0; r < 8; ++r) {
    C[(size_t)(rowBase + r) * NCOLS + n] = acc[r];
  }
}

// ---------------------------------------------------------------------------
// Edge aggregation: one wave per (edge | self-loop).
//   norm = dinv[src] * w * dinv[dst];  out[dst,:] += h[src,:] * norm
// ---------------------------------------------------------------------------
template <int CH>
__global__ void aggregate_kernel(const float* __restrict__ h,
                                 const long long* __restrict__ srcI,
                                 const long long* __restrict__ dstI,
                                 const float* __restrict__ ew,
                                 const float* __restrict__ dinv,
                                 float* __restrict__ out, int E, int N) {
  const int lane = threadIdx.x & 31;
  const int wave = threadIdx.x >> 5;
  const long long item =
      (long long)blockIdx.x * (long long)(blockDim.x >> 5) + wave;
  const long long total = (long long)E + N;
  if (item >= total) return;

  int s, d;
  float w;
  if (item < E) {
    s = (int)srcI[item];
    d = (int)dstI[item];
    w = ew[item];
  } else {
    s = d = (int)(item - E);
    w = 1.0f;  // self loop, fill_value = 1.0
  }
  const float norm = dinv[s] * w * dinv[d];
  const size_t srow = (size_t)s * CH;
  const size_t drow = (size_t)d * CH;
#pragma unroll
  for (int c = lane; c < CH; c += 32) {
    atomicAdd(&out[drow + c], h[srow + c] * norm);
  }
}

// ---------------------------------------------------------------------------
// x = relu(x + bias[ch])
// ---------------------------------------------------------------------------
template <int CH>
__global__ void bias_relu_kernel(float* __restrict__ x,
                                 const float* __restrict__ b, int N) {
  long long i = (long long)blockIdx.x * blockDim.x + threadIdx.x;
  if (i >= (long long)N * CH) return;
  float v = x[i] + b[(int)(i & (CH - 1))];
  x[i] = v > 0.0f ? v : 0.0f;
}

// ---------------------------------------------------------------------------
// Launch
// ---------------------------------------------------------------------------
extern "C" void kernel_launch(void* const* d_in, const int* in_sizes, int n_in,
                              void* d_out, int out_size, void* d_ws,
                              size_t ws_size, hipStream_t stream) {
  const float* x = (const float*)d_in[0];
  const long long* eidx = (const long long*)d_in[1];  // int64 [2, E]
  const float* ew = (const float*)d_in[2];
  const float* W1 = (const float*)d_in[3];
  const float* b1 = (const float*)d_in[4];
  const float* W2 = (const float*)d_in[5];
  const float* b2 = (const float*)d_in[6];
  float* out = (float*)d_out;

  const int N = in_sizes[0] / 64;  // 50000
  const int E = in_sizes[2];       // 800000
  const long long* srcI = eidx;
  const long long* dstI = eidx + E;

  // Workspace carve-out (floats): deg N | dinv N | h1 64N | g1 64N | h2 32N
  float* ws = (float*)d_ws;
  float* deg = ws;
  float* dinv = deg + N;
  float* h1 = dinv + N;
  float* g1 = h1 + (size_t)N * 64;
  float* h2 = g1 + (size_t)N * 64;

  // Zero accumulation buffers (harness poisons d_out/d_ws; must re-zero
  // every call for deterministic graph replay).
  zero_kernel<<<256, 256, 0, stream>>>(deg, N);
  zero_kernel<<<1024, 256, 0, stream>>>(g1, N * 64);
  zero_kernel<<<1024, 256, 0, stream>>>(out, N * 32);

  const int totEN = E + N;
  degree_kernel<<<(totEN + 255) / 256, 256, 0, stream>>>(dstI, ew, deg, E, N);
  dinv_kernel<<<(N + 255) / 256, 256, 0, stream>>>(deg, dinv, N);

  // Layer 1: h1 = x @ W1  (50000x64 @ 64x64), WMMA f32
  {
    int tiles = (N / 16) * (64 / 16);  // 12500
    gemm_wmma_kernel<64><<<(tiles + 3) / 4, 128, 0, stream>>>(x, W1, h1, N);
  }
  aggregate_kernel<64><<<(totEN + 7) / 8, 256, 0, stream>>>(h1, srcI, dstI, ew,
                                                            dinv, g1, E, N);
  bias_relu_kernel<64><<<(int)(((long long)N * 64 + 255) / 256), 256, 0,
                         stream>>>(g1, b1, N);

  // Layer 2: h2 = g1 @ W2  (50000x64 @ 64x32), WMMA f32
  {
    int tiles = (N / 16) * (32 / 16);  // 6250
    gemm_wmma_kernel<32><<<(tiles + 3) / 4, 128, 0, stream>>>(g1, W2, h2, N);
  }
  aggregate_kernel<32><<<(totEN + 7) / 8, 256, 0, stream>>>(h2, srcI, dstI, ew,
                                                            dinv, out, E, N);
  bias_relu_kernel<32><<<(int)(((long long)N * 32 + 255) / 256), 256, 0,
                         stream>>>(out, b2, N);
}